// DetectionLoss_83872121356584
// MI455X (gfx1250) — compile-verified
//
#include <hip/hip_runtime.h>
#include <math.h>

#define NUM_CLASSES 80
#define NB 16
#define NP 8400
#define NG 64
#define EPSF 1e-7f

typedef __attribute__((ext_vector_type(2))) float v2f;
typedef __attribute__((ext_vector_type(8))) float v8f;

// ---------------------------------------------------------------------------
// Wave32 cross-lane sum via V_WMMA_F32_16X16X4_F32 (f32 tensor-core reduce).
// A (16x4): vgpr0 lanes0-15 = K0 rows, lanes16-31 = K2 rows; vgpr1 = K1/K3.
// With a = {p_lane, 0} and B = ones: D[m][n] = p_m + p_{m+16}.
// Lane 0 holds D[0..7][0], lane 16 holds D[8..15][0] -> two partials whose
// sum is the full 32-lane total. Must be called by ALL lanes (EXEC all-1s).
// (clang's atomic optimizer further folds the two lane-atomics into one
//  global_atomic_add_f32 per wave.)
// ---------------------------------------------------------------------------
__device__ __forceinline__ void wave_reduce_atomic(float x, float* dst) {
    v2f a; a[0] = x;    a[1] = 0.0f;
    v2f b; b[0] = 1.0f; b[1] = 1.0f;
    v8f c = {};
    c = __builtin_amdgcn_wmma_f32_16x16x4_f32(false, a, false, b,
                                              (short)0, c, false, false);
    float s = ((c[0] + c[1]) + (c[2] + c[3])) + ((c[4] + c[5]) + (c[6] + c[7]));
    unsigned lane = threadIdx.x & 31u;
    if (lane == 0u || lane == 16u) atomicAdd(dst, s);
}

// bce(l, t=0) = max(l,0) + log1p(exp(-|l|)).
// Use hardware v_exp_f32 / v_log_f32; since exp(-|l|) in (0,1], forming
// 1+x loses <= ~6e-8 absolute vs log1p -- negligible after the mean.
__device__ __forceinline__ float softplus_term(float l) {
    return fmaxf(l, 0.0f) + __logf(1.0f + __expf(-fabsf(l)));
}

__device__ __forceinline__ float nan_to_num_f(float x) {
    if (isnan(x)) return 0.0f;
    if (isinf(x)) return x > 0.0f ? 3.4028235e38f : -3.4028235e38f;
    return x;
}

// ws layout (floats): 0=S_cls  1=S_obj  2=num_fg  3=sum_ciou  4=corr_cls  5=corr_obj
__global__ void init_acc_kernel(float* acc) {
    if (threadIdx.x < 8) acc[threadIdx.x] = 0.0f;
}

// ---------------------------------------------------------------------------
// Kernel 1: assignment-independent softplus sum over all of pred_cls.
// Pure bandwidth: b128 loads, 32-bit grid-stride, WMMA wave reduction.
// ---------------------------------------------------------------------------
__global__ void cls_softplus_kernel(const float4* __restrict__ x4, int n4,
                                    float* __restrict__ acc) {
    const int i = blockIdx.x * blockDim.x + threadIdx.x;
    const int stride = gridDim.x * blockDim.x;
    float s = 0.0f;
    for (int j = i; j < n4; j += stride) {
        float4 v = x4[j];
        s += (softplus_term(v.x) + softplus_term(v.y))
           + (softplus_term(v.z) + softplus_term(v.w));
    }
    wave_reduce_atomic(s, acc + 0);
}

// ---------------------------------------------------------------------------
// Kernel 2: per-point SimOTA-style assignment + CIoU + fg correction terms.
// One block = 256 points of one batch image; GT boxes cached in LDS.
// ---------------------------------------------------------------------------
__global__ void assign_loss_kernel(const float* __restrict__ pred_cls,
                                   const float* __restrict__ pred_obj,
                                   const float* __restrict__ decoded,
                                   const float* __restrict__ points,
                                   const float* __restrict__ strides,
                                   const float* __restrict__ gt_boxes,
                                   const int*   __restrict__ gt_labels,
                                   float* __restrict__ acc) {
    __shared__ float s_x1[NG], s_y1[NG], s_x2[NG], s_y2[NG], s_area[NG];
    __shared__ int   s_lab[NG];

    const int b = blockIdx.y;
    const int p = blockIdx.x * blockDim.x + threadIdx.x;

    if (threadIdx.x < NG) {
        int g = threadIdx.x;
        float x1 = gt_boxes[((size_t)b * NG + g) * 4 + 0];
        float y1 = gt_boxes[((size_t)b * NG + g) * 4 + 1];
        float x2 = gt_boxes[((size_t)b * NG + g) * 4 + 2];
        float y2 = gt_boxes[((size_t)b * NG + g) * 4 + 3];
        s_x1[g] = x1; s_y1[g] = y1; s_x2[g] = x2; s_y2[g] = y2;
        s_area[g] = (x2 - x1) * (y2 - y1);
        s_lab[g]  = gt_labels[(size_t)b * NG + g];
    }
    __syncthreads();

    float a_nfg = 0.0f, a_ciou = 0.0f, a_ccls = 0.0f, a_cobj = 0.0f, a_sobj = 0.0f;

    if (p < NP) {
        const float px = points[2 * p + 0];
        const float py = points[2 * p + 1];
        const float st = strides[p];
        const float rad = st * 2.5f;       // CENTER_RADIUS
        const float fit_thr = st * 8.0f;

        float bestA_f = __builtin_inff(); int bestI_f = 0;   // match (fits)
        float bestA_b = __builtin_inff(); int bestI_b = 0;   // fallback
        for (int g = 0; g < NG; ++g) {
            float gx1 = s_x1[g], gy1 = s_y1[g], gx2 = s_x2[g], gy2 = s_y2[g];
            float d1 = px - gx1, d2 = py - gy1, d3 = gx2 - px, d4 = gy2 - py;
            float dmin = fminf(fminf(d1, d2), fminf(d3, d4));
            float dmax = fmaxf(fmaxf(d1, d2), fmaxf(d3, d4));
            float cx = (gx1 + gx2) * 0.5f, cy = (gy1 + gy2) * 0.5f;
            float e1 = px - (cx - rad), e2 = py - (cy - rad);
            float e3 = cx + rad - px,  e4 = cy + rad - py;
            float emin = fminf(fminf(e1, e2), fminf(e3, e4));
            bool fallback = (dmin > 0.0f) && (emin > 0.0f);
            bool match    = fallback && (dmax <= fit_thr);
            float ar = s_area[g];
            if (match    && ar < bestA_f) { bestA_f = ar; bestI_f = g; }
            if (fallback && ar < bestA_b) { bestA_b = ar; bestI_b = g; }
        }
        bool any_f = isfinite(bestA_f);
        int  idx   = any_f ? bestI_f : bestI_b;
        bool fg    = any_f || isfinite(bestA_b);

        const size_t bp = (size_t)b * NP + p;
        const float lobj = pred_obj[bp];
        a_sobj = softplus_term(lobj);            // obj BCE t=0 part, all points

        if (fg) {
            a_nfg = 1.0f;
            // clean predicted box: nan->0, +inf->1000, -inf->0, clip, order
            float bx[4];
            #pragma unroll
            for (int k = 0; k < 4; ++k) {
                float v = decoded[bp * 4 + k];
                if (isnan(v)) v = 0.0f;
                else if (isinf(v)) v = (v > 0.0f) ? 1000.0f : 0.0f;
                bx[k] = fminf(fmaxf(v, -1000.0f), 1000.0f);
            }
            float x1 = bx[0], y1 = bx[1];
            float x2 = fmaxf(x1 + 0.001f, bx[2]);
            float y2 = fmaxf(y1 + 0.001f, bx[3]);
            float gx1 = s_x1[idx], gy1 = s_y1[idx];
            float gx2 = s_x2[idx], gy2 = s_y2[idx];

            float w1 = x2 - x1,  h1 = y2 - y1;
            float w2 = gx2 - gx1, h2 = gy2 - gy1;
            float iw = fmaxf(fminf(x2, gx2) - fmaxf(x1, gx1), 0.0f);
            float ih = fmaxf(fminf(y2, gy2) - fmaxf(y1, gy1), 0.0f);
            float inter = iw * ih;
            float uni = w1 * h1 + w2 * h2 - inter + EPSF;
            float iou = inter / uni;
            float cw = fmaxf(x2, gx2) - fminf(x1, gx1);
            float ch = fmaxf(y2, gy2) - fminf(y1, gy1);
            float cc = cw * cw + ch * ch + EPSF;
            float ddx = x1 + x2 - gx1 - gx2;
            float ddy = y1 + y2 - gy1 - gy2;
            float rho2 = (ddx * ddx + ddy * ddy) * 0.25f;
            float da = atanf(w2 / (h2 + EPSF)) - atanf(w1 / (h1 + EPSF));
            float vv = (4.0f / 9.869604401089358f) * da * da;   // 4/pi^2
            float alpha = vv / (1.0f - iou + vv + EPSF);
            float ci = 1.0f - (iou - rho2 / cc - alpha * vv);
            if (isnan(ci) || isinf(ci)) ci = 1.0f;
            ci = fminf(fmaxf(ci, 0.0f), 2.0f);
            a_ciou = ci;

            float t = fminf(fmaxf(1.0f - ci, 0.0f), 1.0f);      // obj target
            a_cobj = lobj * t;
            a_ccls = pred_cls[bp * NUM_CLASSES + s_lab[idx]];
        }
    }

    // All lanes participate (EXEC all-1s required by WMMA).
    wave_reduce_atomic(a_sobj, acc + 1);
    wave_reduce_atomic(a_nfg,  acc + 2);
    wave_reduce_atomic(a_ciou, acc + 3);
    wave_reduce_atomic(a_ccls, acc + 4);
    wave_reduce_atomic(a_cobj, acc + 5);
}

// ---------------------------------------------------------------------------
// Kernel 3: combine accumulators into the 4 output losses.
// ---------------------------------------------------------------------------
__global__ void finalize_kernel(const float* __restrict__ acc,
                                float* __restrict__ out) {
    const float S_cls = acc[0], S_obj = acc[1], nfg = acc[2];
    const float sum_ciou = acc[3], corr_cls = acc[4], corr_obj = acc[5];

    float loss_cls = (S_cls - corr_cls) / (float)((size_t)NB * NP * NUM_CLASSES);
    loss_cls = fminf(nan_to_num_f(loss_cls), 10.0f);

    float loss_obj = (S_obj - corr_obj) / (float)((size_t)NB * NP);
    loss_obj = fminf(nan_to_num_f(loss_obj), 10.0f);

    float normalizer = fmaxf(nfg, 1.0f);
    float loss_box = fminf(nan_to_num_f(sum_ciou / normalizer), 10.0f);

    const float WARMUP = 1.0f / 3.0f;   // min(1, (0+1)/3)
    float total = 0.5f * loss_cls + 7.5f * WARMUP * loss_box + 1.0f * loss_obj;
    total = fminf(nan_to_num_f(total), 10.0f);

    out[0] = total; out[1] = loss_cls; out[2] = loss_box; out[3] = loss_obj;
}

extern "C" void kernel_launch(void* const* d_in, const int* in_sizes, int n_in,
                              void* d_out, int out_size, void* d_ws, size_t ws_size,
                              hipStream_t stream) {
    const float* pred_cls = (const float*)d_in[0];
    // d_in[1] = pred_box (unused by the reference)
    const float* pred_obj = (const float*)d_in[2];
    const float* decoded  = (const float*)d_in[3];
    const float* points   = (const float*)d_in[4];
    const float* strides  = (const float*)d_in[5];
    const float* gt_boxes = (const float*)d_in[6];
    const int*   gt_labels= (const int*)  d_in[7];
    float* acc = (float*)d_ws;
    float* out = (float*)d_out;

    init_acc_kernel<<<1, 32, 0, stream>>>(acc);

    const int n4 = NB * NP * NUM_CLASSES / 4;   // 2,688,000
    cls_softplus_kernel<<<1024, 256, 0, stream>>>(
        (const float4*)pred_cls, n4, acc);

    dim3 grid((NP + 255) / 256, NB);
    assign_loss_kernel<<<grid, 256, 0, stream>>>(
        pred_cls, pred_obj, decoded, points, strides, gt_boxes, gt_labels, acc);

    finalize_kernel<<<1, 1, 0, stream>>>(acc, out);
}